// GCRN_28114855919849
// MI455X (gfx1250) — compile-verified
//
#include <hip/hip_runtime.h>
#include <stdint.h>

// ---------------------------------------------------------------------------
// GConvGRU (K=2 Chebyshev) with H0 = 0  =>  algebraic collapse:
//   Z  = sigmoid(x@Wxz0 + TX1@Wxz1 + bxz + bhz)
//   Ht = tanh   (x@Wxh0 + TX1@Wxh1 + bxh + bhh)
//   out = relu((1-Z)*Ht) @ linW + lin_b
// where TX1[dst] += (-dinv[src]*w*dinv[dst]) * x[src], dinv = rsqrt(segsum_src w).
// R-gate is dead code (multiplied by H0 = 0).
//
// MI455X plan: edge phase is L2-resident scatter (x and TX1 are 12.8 MB each,
// L2 = 192 MB) -> memory bound, ~tens of us. Dense phase is only 1.6 GFLOP so
// we keep full f32 precision with V_WMMA_F32_16X16X4_F32 (one 16-node tile
// per wave32; 256 wmma per tile).
// ---------------------------------------------------------------------------

#define NN 50000
#define NE 800000
#define CH 64
#define NTILES (NN / 16)        // 3125 exactly
#define WAVES_PER_BLOCK 8

typedef __attribute__((ext_vector_type(2))) float v2f;
typedef __attribute__((ext_vector_type(8))) float v8f;

// ---- Pass 1: degree by src --------------------------------------------------
__global__ void gcrn_deg(const long long* __restrict__ ei,
                         const float* __restrict__ w,
                         float* __restrict__ deg) {
    int e = blockIdx.x * blockDim.x + threadIdx.x;
    if (e < NE) atomicAdd(&deg[(int)ei[e]], w[e]);
}

// ---- Pass 2: dinv -----------------------------------------------------------
__global__ void gcrn_dinv(const float* __restrict__ deg,
                          float* __restrict__ dinv) {
    int n = blockIdx.x * blockDim.x + threadIdx.x;
    if (n < NN) {
        float d = deg[n];
        dinv[n] = (d > 0.0f) ? rsqrtf(d) : 0.0f;
    }
}

// ---- Pass 3: TX1[dst] += norm * x[src]  (16 threads/edge, float4 each) ------
__global__ void gcrn_scatter(const long long* __restrict__ ei,
                             const float* __restrict__ w,
                             const float* __restrict__ dinv,
                             const float* __restrict__ x,
                             float* __restrict__ tx1) {
    int gid = blockIdx.x * blockDim.x + threadIdx.x;
    int e   = gid >> 4;
    if (e >= NE) return;
    int c4  = (gid & 15) * 4;
    int s   = (int)ei[e];
    int d   = (int)ei[NE + e];
    float nrm = -dinv[s] * w[e] * dinv[d];
    const float4 xv = *(const float4*)(x + (size_t)s * CH + c4);
    float* o = tx1 + (size_t)d * CH + c4;
    atomicAdd(o + 0, nrm * xv.x);
    atomicAdd(o + 1, nrm * xv.y);
    atomicAdd(o + 2, nrm * xv.z);
    atomicAdd(o + 3, nrm * xv.w);
}

// ---- WMMA fragment helpers (V_WMMA_F32_16X16X4_F32 layouts) ----------------
// A 16x4: lane l<16 holds A[l][k0],A[l][k0+1]; lane l+16 holds A[l][k0+2..3].
__device__ __forceinline__ v2f frag_a(const float* __restrict__ base,
                                      int row, int k0, int hi) {
    const float* p = base + (size_t)row * CH + k0 + 2 * hi;
    v2f a; a.x = p[0]; a.y = p[1];
    return a;
}
// B 4x16 (row-major W[k][n]): lane holds W[k0+2*hi][col], W[k0+2*hi+1][col].
__device__ __forceinline__ v2f frag_b(const float* __restrict__ W,
                                      int k0, int hi, int col) {
    int k = k0 + 2 * hi;
    v2f b; b.x = W[k * CH + col]; b.y = W[(k + 1) * CH + col];
    return b;
}

// ---- Pass 4: fused node transform + GRU activations + regression head ------
__global__ void __launch_bounds__(32 * WAVES_PER_BLOCK)
gcrn_node(const float* __restrict__ x,  const float* __restrict__ tx1,
          const float* __restrict__ Wz, // Wx_z [2,64,64]
          const float* __restrict__ bxz, const float* __restrict__ bhz,
          const float* __restrict__ Wh, // Wx_h [2,64,64]
          const float* __restrict__ bxh, const float* __restrict__ bhh,
          const float* __restrict__ linW, const float* __restrict__ linb,
          float* __restrict__ out) {
    const int lane = threadIdx.x & 31;
    const int wave = threadIdx.x >> 5;
    const int tile = blockIdx.x * WAVES_PER_BLOCK + wave;
    if (tile >= NTILES) return;          // wave-uniform: EXEC all-1s inside

    const int nb = tile * 16;
    const int m  = lane & 15;            // A row / BCD column within tile
    const int hi = lane >> 4;

    const float* Wz0 = Wz;
    const float* Wz1 = Wz + CH * CH;
    const float* Wh0 = Wh;
    const float* Wh1 = Wh + CH * CH;

    v8f accZ[4], accH[4];
#pragma unroll
    for (int t = 0; t < 4; ++t) {
        float bz = bxz[t * 16 + m] + bhz[t * 16 + m];   // H0=0: Wh terms = bias
        float bh = bxh[t * 16 + m] + bhh[t * 16 + m];
#pragma unroll
        for (int v = 0; v < 8; ++v) { accZ[t][v] = bz; accH[t][v] = bh; }
    }

    for (int s = 0; s < 16; ++s) {       // K = 64 in steps of 4
        const int k0 = 4 * s;
        v2f ax = frag_a(x,   nb + m, k0, hi);
        v2f at = frag_a(tx1, nb + m, k0, hi);
#pragma unroll
        for (int t = 0; t < 4; ++t) {    // N = 64 in tiles of 16
            const int col = t * 16 + m;
            accZ[t] = __builtin_amdgcn_wmma_f32_16x16x4_f32(
                false, ax, false, frag_b(Wz0, k0, hi, col), (short)0, accZ[t], false, false);
            accZ[t] = __builtin_amdgcn_wmma_f32_16x16x4_f32(
                false, at, false, frag_b(Wz1, k0, hi, col), (short)0, accZ[t], false, false);
            accH[t] = __builtin_amdgcn_wmma_f32_16x16x4_f32(
                false, ax, false, frag_b(Wh0, k0, hi, col), (short)0, accH[t], false, false);
            accH[t] = __builtin_amdgcn_wmma_f32_16x16x4_f32(
                false, at, false, frag_b(Wh1, k0, hi, col), (short)0, accH[t], false, false);
        }
    }

    // Activations + fold linW into per-lane partials.
    float p[8];
#pragma unroll
    for (int v = 0; v < 8; ++v) p[v] = 0.0f;
#pragma unroll
    for (int t = 0; t < 4; ++t) {
        float lw = linW[t * 16 + m];
#pragma unroll
        for (int v = 0; v < 8; ++v) {
            float z  = 1.0f / (1.0f + expf(-accZ[t][v]));
            float ht = tanhf(accH[t][v]);
            float Hv = (1.0f - z) * ht;
            float r  = Hv > 0.0f ? Hv : 0.0f;
            p[v] += r * lw;
        }
    }
    // Row v+8*hi lives across the 16 lanes of each wave half -> xor-tree reduce.
#pragma unroll
    for (int off = 8; off >= 1; off >>= 1)
#pragma unroll
        for (int v = 0; v < 8; ++v)
            p[v] += __shfl_xor(p[v], off, 16);

    if (m == 0) {
        float lb = linb[0];
#pragma unroll
        for (int v = 0; v < 8; ++v)
            out[nb + hi * 8 + v] = p[v] + lb;
    }
}

// ---------------------------------------------------------------------------
extern "C" void kernel_launch(void* const* d_in, const int* in_sizes, int n_in,
                              void* d_out, int out_size, void* d_ws, size_t ws_size,
                              hipStream_t stream) {
    const float*     x    = (const float*)d_in[0];
    const long long* ei   = (const long long*)d_in[1];   // int64 edge_index
    const float*     w    = (const float*)d_in[2];
    const float*     Wx_z = (const float*)d_in[3];
    const float*     bx_z = (const float*)d_in[4];
    const float*     bh_z = (const float*)d_in[6];
    // R-gate (indices 7..10) provably dead: multiplied by H0 == 0.
    const float*     Wx_h = (const float*)d_in[11];
    const float*     bx_h = (const float*)d_in[12];
    const float*     bh_h = (const float*)d_in[14];
    const float*     linW = (const float*)d_in[15];
    const float*     linb = (const float*)d_in[16];
    float*           out  = (float*)d_out;

    float* deg  = (float*)d_ws;          // [NN]
    float* dinv = deg + NN;              // [NN]
    float* tx1  = dinv + NN;             // [NN*CH]

    hipMemsetAsync(deg, 0, (size_t)NN * sizeof(float), stream);
    hipMemsetAsync(tx1, 0, (size_t)NN * CH * sizeof(float), stream);

    gcrn_deg <<<(NE + 255) / 256, 256, 0, stream>>>(ei, w, deg);
    gcrn_dinv<<<(NN + 255) / 256, 256, 0, stream>>>(deg, dinv);
    gcrn_scatter<<<(NE * 16) / 256, 256, 0, stream>>>(ei, w, dinv, x, tx1);
    gcrn_node<<<(NTILES + WAVES_PER_BLOCK - 1) / WAVES_PER_BLOCK,
                32 * WAVES_PER_BLOCK, 0, stream>>>(
        x, tx1, Wx_z, bx_z, bh_z, Wx_h, bx_h, bh_h, linW, linb, out);
}